// FeatureMatchSimpleLoss_68633577390702
// MI455X (gfx1250) — compile-verified
//
#include <hip/hip_runtime.h>
#include <hip/hip_bf16.h>
#include <math.h>

typedef __attribute__((ext_vector_type(2))) float v2f;
typedef __attribute__((ext_vector_type(8))) float v8f;

#define B_IMG 128
#define P_PTS 512
#define D_DIM 384
#define GAMMA 20
#define LAMBDA_INV 25.0f

#define MBLK 128              // rows per block (8 waves x 16 rows)
#define KC 32                 // K-chunk (8 wmma steps of K=4)
#define LDA 36                // LDS row stride (floats): 16B-aligned, conflict-spread
#define NCHUNK (D_DIM / KC)   // 12
#define NTG 4                 // column tiles per group (4 accumulators / wave)
#define NGRP (P_PTS / (16 * NTG))  // 8 groups of 64 columns
#define NTOT (B_IMG * P_PTS)  // 65536

// ---- CDNA5 async global->LDS path (guarded; falls back to ld/st) ----------
#if defined(__has_builtin)
#if __has_builtin(__builtin_amdgcn_global_load_async_to_lds_b128)
#define HAVE_ASYNC_LDS 1
#endif
#endif

#if defined(HAVE_ASYNC_LDS)
// Toolchain signature (from diagnostics): params are pointers to
// `int __attribute__((vector_size(16)))` in the generic address space.
typedef int v4i_vs __attribute__((vector_size(16)));
#endif

__device__ __forceinline__ void copy16(const float* __restrict__ gp,
                                       float* __restrict__ lp) {
#if defined(HAVE_ASYNC_LDS)
  __builtin_amdgcn_global_load_async_to_lds_b128((v4i_vs*)gp, (v4i_vs*)lp, 0, 0);
#else
  *reinterpret_cast<float4*>(lp) = *reinterpret_cast<const float4*>(gp);
#endif
}

__device__ __forceinline__ void wait_async0() {
#if defined(HAVE_ASYNC_LDS)
#if __has_builtin(__builtin_amdgcn_s_wait_asynccnt)
  __builtin_amdgcn_s_wait_asynccnt(0);
#else
  asm volatile("s_wait_asynccnt 0" ::: "memory");
#endif
#endif
}

// ---------------------------------------------------------------------------
// Kernel 1: per-image Gram matrix via V_WMMA_F32_16X16X4_F32 with masked
// row max/argmax.  Each wave owns 16 rows; 4 column tiles per K-sweep.
// ---------------------------------------------------------------------------
__device__ __forceinline__ void load_chunk(const float* __restrict__ zb,
                                           float* __restrict__ sAbuf,
                                           float* __restrict__ sBbuf,
                                           int m0, int g, int k0, int tid) {
  // A: MBLK x KC floats = 1024 float4, 256 threads -> 4 each
#pragma unroll
  for (int f = tid; f < (MBLK * KC) / 4; f += 256) {
    const int r = f / (KC / 4);
    const int k4 = (f % (KC / 4)) * 4;
    copy16(zb + (size_t)(m0 + r) * D_DIM + k0 + k4, &sAbuf[r * LDA + k4]);
  }
  // B: 64 x KC floats = 512 float4, 2 each
#pragma unroll
  for (int f = tid; f < (64 * KC) / 4; f += 256) {
    const int r = f / (KC / 4);
    const int k4 = (f % (KC / 4)) * 4;
    copy16(zb + (size_t)(g * 64 + r) * D_DIM + k0 + k4, &sBbuf[r * LDA + k4]);
  }
}

__device__ __forceinline__ void mma_chunk(const float* __restrict__ Abuf,
                                          const float* __restrict__ Bbuf,
                                          int arow, int l15, int half, v8f c[NTG]) {
#pragma unroll
  for (int kk = 0; kk < KC; kk += 4) {
    const float2 af = *reinterpret_cast<const float2*>(&Abuf[arow * LDA + kk + 2 * half]);
    v2f av; av.x = af.x; av.y = af.y;
#pragma unroll
    for (int t = 0; t < NTG; ++t) {
      const float2 bf =
          *reinterpret_cast<const float2*>(&Bbuf[(t * 16 + l15) * LDA + kk + 2 * half]);
      v2f bv; bv.x = bf.x; bv.y = bf.y;
      c[t] = __builtin_amdgcn_wmma_f32_16x16x4_f32(false, av, false, bv, (short)0,
                                                   c[t], false, false);
    }
  }
}

__global__ __launch_bounds__(256) void fmatch_sim_kernel(
    const float* __restrict__ z, const int* __restrict__ view_ids,
    float* __restrict__ best_sim, int* __restrict__ best_j) {
  __shared__ float sA[2][MBLK * LDA];
  __shared__ float sB[2][64 * LDA];
  __shared__ int sview[P_PTS];

  const int tid = threadIdx.x;
  const int wave = tid >> 5;
  const int lane = tid & 31;
  const int half = lane >> 4;
  const int l15 = lane & 15;

  const int b = blockIdx.x / (P_PTS / MBLK);
  const int m0 = (blockIdx.x % (P_PTS / MBLK)) * MBLK;
  const float* zb = z + (size_t)b * P_PTS * D_DIM;

  for (int i = tid; i < P_PTS; i += 256) sview[i] = view_ids[i];
  __syncthreads();

  const int arow = wave * 16 + l15;  // row within block strip (A fragment)

  // per-lane row metadata: vgpr e holds rows wave*16 + 8*half + e
  int prow[8];
  int pview[8];
#pragma unroll
  for (int e = 0; e < 8; ++e) {
    prow[e] = m0 + wave * 16 + 8 * half + e;  // row index within image
    pview[e] = sview[prow[e]];
  }

  float best[8];
  int bestj[8];
#pragma unroll
  for (int e = 0; e < 8; ++e) { best[e] = -INFINITY; bestj[e] = 0x7fffffff; }

  for (int g = 0; g < NGRP; ++g) {
    v8f c[NTG];
#pragma unroll
    for (int t = 0; t < NTG; ++t) c[t] = (v8f){0.f, 0.f, 0.f, 0.f, 0.f, 0.f, 0.f, 0.f};

    load_chunk(zb, sA[0], sB[0], m0, g, 0, tid);
    wait_async0();
    __syncthreads();
    for (int kc = 0; kc < NCHUNK; ++kc) {
      if (kc + 1 < NCHUNK)
        load_chunk(zb, sA[(kc + 1) & 1], sB[(kc + 1) & 1], m0, g, (kc + 1) * KC, tid);
      mma_chunk(sA[kc & 1], sB[kc & 1], arow, l15, half, c);
      wait_async0();
      __syncthreads();
    }

    // masked row-max update; invalid entries behave as value -1.0 (index kept)
#pragma unroll
    for (int t = 0; t < NTG; ++t) {
      const int q = g * 64 + t * 16 + l15;  // column within image (fixed per lane)
      const int qview = sview[q];
#pragma unroll
      for (int e = 0; e < 8; ++e) {
        const bool valid = (qview != pview[e]) && (q != prow[e]);
        const float v = valid ? c[t][e] : -1.0f;
        if (v > best[e]) { best[e] = v; bestj[e] = q; }  // strict >: lowest q wins ties
      }
    }
  }

  // reduce across the 16 lanes of each half (xor masks 1,2,4,8 stay in-half)
#pragma unroll
  for (int e = 0; e < 8; ++e) {
#pragma unroll
    for (int m = 1; m <= 8; m <<= 1) {
      const float ov = __shfl_xor(best[e], m, 32);
      const int oj = __shfl_xor(bestj[e], m, 32);
      if (ov > best[e] || (ov == best[e] && oj < bestj[e])) {
        best[e] = ov;
        bestj[e] = oj;
      }
    }
    if (l15 == 0) {
      const int gidx = b * P_PTS + prow[e];
      best_sim[gidx] = best[e];
      best_j[gidx] = b * P_PTS + bestj[e];  // global flat match index
    }
  }
}

// ---------------------------------------------------------------------------
// Kernel 2: global top-GAMMA (stable, lowest-index ties) + final loss.
// ---------------------------------------------------------------------------
__global__ __launch_bounds__(256) void fmatch_topk_loss_kernel(
    const float* __restrict__ z, float* __restrict__ best_sim,
    const int* __restrict__ best_j, float* __restrict__ out) {
  __shared__ float sv[256];
  __shared__ int si[256];
  __shared__ int chosen[GAMMA];
  __shared__ float chosenV[GAMMA];
  const int tid = threadIdx.x;

  for (int r = 0; r < GAMMA; ++r) {
    float bv = -INFINITY;
    int bi = 0x7fffffff;
    for (int i = tid; i < NTOT; i += 256) {
      const float v = best_sim[i];
      if (v > bv) { bv = v; bi = i; }
    }
    sv[tid] = bv;
    si[tid] = bi;
    __syncthreads();
    for (int s = 128; s > 0; s >>= 1) {
      if (tid < s) {
        const float ov = sv[tid + s];
        const int oi = si[tid + s];
        if (ov > sv[tid] || (ov == sv[tid] && oi < si[tid])) {
          sv[tid] = ov;
          si[tid] = oi;
        }
      }
      __syncthreads();
    }
    if (tid == 0) {
      chosen[r] = si[0];
      chosenV[r] = sv[0];
      best_sim[si[0]] = -INFINITY;  // exclude from later rounds (k1 rewrites each call)
    }
    __syncthreads();
  }

  // loss = LAMBDA_INV * mean((z1 - z2)^2) over [GAMMA, D]
  float part = 0.f;
  for (int r = 0; r < GAMMA; ++r) {
    const int i1 = chosen[r];
    const int i2 = best_j[i1];
    const float* p1 = z + (size_t)i1 * D_DIM;
    const float* p2 = z + (size_t)i2 * D_DIM;
    for (int d = tid; d < D_DIM; d += 256) {
      const float df = p1[d] - p2[d];
      part += df * df;
    }
  }
  sv[tid] = part;
  __syncthreads();
  for (int s = 128; s > 0; s >>= 1) {
    if (tid < s) sv[tid] += sv[tid + s];
    __syncthreads();
  }
  if (tid == 0) {
    out[0] = LAMBDA_INV * sv[0] / (float)(GAMMA * D_DIM);
    float m = 0.f;
    for (int r = 0; r < GAMMA; ++r) m += chosenV[r];
    out[1] = m / (float)GAMMA;
  }
}

extern "C" void kernel_launch(void* const* d_in, const int* in_sizes, int n_in,
                              void* d_out, int out_size, void* d_ws, size_t ws_size,
                              hipStream_t stream) {
  const float* z = (const float*)d_in[0];
  const int* view_ids = (const int*)d_in[1];
  float* out = (float*)d_out;

  float* best_sim = (float*)d_ws;                            // NTOT floats
  int* best_j = (int*)((char*)d_ws + NTOT * sizeof(float));  // NTOT ints

  const int blocks = B_IMG * (P_PTS / MBLK);  // 512
  fmatch_sim_kernel<<<blocks, 256, 0, stream>>>(z, view_ids, best_sim, best_j);
  fmatch_topk_loss_kernel<<<1, 256, 0, stream>>>(z, best_sim, best_j, out);
}